// linearGELU2_52295521796802
// MI455X (gfx1250) — compile-verified
//
#include <hip/hip_runtime.h>

// Native clang vector (NOT HIP_vector_type) so __builtin_nontemporal_* accepts it.
typedef float v4f __attribute__((ext_vector_type(4)));

// Piecewise-linear GELU2, faithful to the reference (x<-3 and x>=3 pass
// through unchanged). Branchless: each range gate becomes v_cmp + v_cndmask,
// EXEC stays fully dense (good for wave32 issue).
__device__ __forceinline__ float lgelu2(float x) {
    float y = x;
    // [-3, -1): -0.0773*(x+3) - 0.004
    {
        float e = -0.0773f * (x + 3.0f) - 0.004f;
        y = (x >= -3.0f && x < -1.0f) ? e : y;
    }
    // [-1, 0): 0.1587*(x+1) - 0.1587
    {
        float e = 0.1587f * (x + 1.0f) - 0.1587f;
        y = (x >= -1.0f && x < 0.0f) ? e : y;
    }
    // [0, 1): 0.8413*x
    {
        float e = 0.8413f * x;
        y = (x >= 0.0f && x < 1.0f) ? e : y;
    }
    // [1, 3): 1.0773*(x-1) + 0.8413
    {
        float e = 1.0773f * (x - 1.0f) + 0.8413f;
        y = (x >= 1.0f && x < 3.0f) ? e : y;
    }
    return y;
}

__device__ __forceinline__ v4f lgelu2_v4(v4f v) {
    v4f r;
    r.x = lgelu2(v.x);
    r.y = lgelu2(v.y);
    r.z = lgelu2(v.z);
    r.w = lgelu2(v.w);
    return r;
}

// Exact-tile streaming kernel: launcher guarantees n4 == gridDim*blockDim*4.
// Each thread issues 4 independent 128-bit NT loads back-to-back (deep MLP,
// clause-able), computes, then 4 NT stores. All accesses wave-coalesced.
// Touch-once data (512 MB >> 192 MB L2) -> NT hints keep L2 clean.
__global__ void linearGELU2_exact_kernel(const v4f* __restrict__ in,
                                         v4f* __restrict__ out) {
    const int stride = gridDim.x * blockDim.x;
    const int i = blockIdx.x * blockDim.x + threadIdx.x;

    v4f a = __builtin_nontemporal_load(&in[i]);              // global_load_b128 NT
    v4f b = __builtin_nontemporal_load(&in[i + stride]);
    v4f c = __builtin_nontemporal_load(&in[i + 2 * stride]);
    v4f d = __builtin_nontemporal_load(&in[i + 3 * stride]);

    a = lgelu2_v4(a);
    b = lgelu2_v4(b);
    c = lgelu2_v4(c);
    d = lgelu2_v4(d);

    __builtin_nontemporal_store(a, &out[i]);                 // global_store_b128 NT
    __builtin_nontemporal_store(b, &out[i + stride]);
    __builtin_nontemporal_store(c, &out[i + 2 * stride]);
    __builtin_nontemporal_store(d, &out[i + 3 * stride]);
}

// Generic grid-stride fallback (any n4), with gfx1250 global_prefetch_b8 for
// the next line this thread will touch.
__global__ void linearGELU2_generic_kernel(const v4f* __restrict__ in,
                                           v4f* __restrict__ out,
                                           int n4) {
    int i      = blockIdx.x * blockDim.x + threadIdx.x;
    int stride = gridDim.x * blockDim.x;
    for (; i < n4; i += stride) {
        __builtin_prefetch(&in[i + stride], 0, 1);           // global_prefetch_b8
        v4f v = __builtin_nontemporal_load(&in[i]);
        __builtin_nontemporal_store(lgelu2_v4(v), &out[i]);
    }
}

// Scalar tail kernel in case the flat size isn't a multiple of 4.
__global__ void linearGELU2_tail_kernel(const float* __restrict__ in,
                                        float* __restrict__ out,
                                        int n) {
    int i = blockIdx.x * blockDim.x + threadIdx.x;
    if (i < n) out[i] = lgelu2(in[i]);
}

extern "C" void kernel_launch(void* const* d_in, const int* in_sizes, int n_in,
                              void* d_out, int out_size, void* d_ws, size_t ws_size,
                              hipStream_t stream) {
    (void)n_in; (void)d_ws; (void)ws_size; (void)out_size;
    const float* x = (const float*)d_in[0];
    float* y       = (float*)d_out;
    const long long n = (long long)in_sizes[0];   // 16*4096*1024 = 67,108,864

    const int threads = 256;                      // 8 wave32 waves / block
    const int unroll  = 4;                        // v4f per thread (exact kernel)

    if ((n & 3LL) == 0LL) {
        const int n4 = (int)(n >> 2);             // 16,777,216 float4s
        const long long tile = (long long)threads * unroll;
        if ((n4 % tile) == 0) {
            int blocks = (int)(n4 / tile);        // 16384 for this problem
            linearGELU2_exact_kernel<<<blocks, threads, 0, stream>>>(
                (const v4f*)x, (v4f*)y);
        } else {
            int blocks = (n4 + threads * unroll - 1) / (threads * unroll);
            linearGELU2_generic_kernel<<<blocks, threads, 0, stream>>>(
                (const v4f*)x, (v4f*)y, n4);
        }
    } else {
        int blocks = (int)((n + threads - 1) / threads);
        linearGELU2_tail_kernel<<<blocks, threads, 0, stream>>>(x, y, (int)n);
    }
}